// DiceBCELossWithTopology_34668976013761
// MI455X (gfx1250) — compile-verified
//
#include <hip/hip_runtime.h>
#include <hip/hip_bf16.h>
#include <stdint.h>

// ---------------------------------------------------------------------------
// DiceBCELossWithTopology fused reduction for MI455X (gfx1250).
//
// Memory-bound: 134 MB streamed once -> ~5.8 us at 23.3 TB/s. No WMMA (no
// matrix structure); the CDNA5 path that matters is async global->LDS
// staging (ASYNCcnt) with double buffering, wave32 blocks, f32 atomics.
// ---------------------------------------------------------------------------

#define N_ELEMS   16777216            // 64*1*512*512
#define NBLOCKS   2048
#define TPB       256
#define VEC       4
#define CHUNK     (TPB * VEC)         // 1024 elements per chunk
#define PER_BLOCK (N_ELEMS / NBLOCKS) // 8192 elements per block (contiguous)
#define NCHUNKS   (PER_BLOCK / CHUNK) // 8 chunks
#define IMG_PIX   262144.0f           // 512*512
#define LOG_CLAMP (-100.0f)

#if defined(__HIP_DEVICE_COMPILE__)

#define AS1 __attribute__((address_space(1)))
#define AS3 __attribute__((address_space(3)))
typedef __attribute__((ext_vector_type(4))) int v4i;

__device__ __forceinline__ void async_load_b128(const float* g, float* l) {
#if __has_builtin(__builtin_amdgcn_global_load_async_to_lds_b128)
  // Probe-confirmed (round 1 stderr): param 0 is 'int4 __device__ *'
  // (addrspace(1) pointer to 16-byte vector).
  __builtin_amdgcn_global_load_async_to_lds_b128(
      (AS1 v4i*)(uintptr_t)g,
      (AS3 v4i*)(unsigned)(uintptr_t)l,
      0, 0);
#else
  // GV mode: VDST = LDS byte offset, VADDR = 64-bit global address.
  asm volatile("global_load_async_to_lds_b128 %0, %1, off"
               :: "v"((unsigned)(uintptr_t)l), "v"(g)
               : "memory");
#endif
}

#if __has_builtin(__builtin_amdgcn_s_wait_asynccnt)
#define WAIT_ASYNC(n) __builtin_amdgcn_s_wait_asynccnt(n)
#else
#define WAIT_ASYNC(n) asm volatile("s_wait_asynccnt " #n ::: "memory")
#endif

#else  // host pass: stubs so sema never sees target builtins/asm
__device__ __forceinline__ void async_load_b128(const float* g, float* l) { (void)g; (void)l; }
#define WAIT_ASYNC(n) ((void)0)
#endif

// ws layout (floats): [0]=sum_x [1]=sum_t [2]=sum_xt [3]=sum_bce [4]=ncomp(u32)
__global__ void dicebce_init_ws(float* acc) {
  if (threadIdx.x < 8) acc[threadIdx.x] = 0.0f;
}

__global__ __launch_bounds__(TPB)
void dicebce_main(const float* __restrict__ xg, const float* __restrict__ tg,
                  float* __restrict__ acc) {
  __shared__ __align__(16) float xbuf[2][CHUNK];
  __shared__ __align__(16) float tbuf[2][CHUNK];
  __shared__ float carry[2];
  __shared__ float red0[TPB], red1[TPB], red2[TPB], red3[TPB];
  __shared__ unsigned redc[TPB];

  const int tid = threadIdx.x;
  const size_t base = (size_t)blockIdx.x * PER_BLOCK;

  // Seed the left-neighbor carry for the run-start (connected components)
  // test. For the very first element of the array, "previous" is treated as
  // zero so run_starts[0] == (x[0] != 0), matching the reference.
  if (tid == 0) carry[0] = (blockIdx.x > 0) ? xg[base - 1] : 0.0f;

  // Prologue: stage chunk 0 into buffer 0 (2 async loads per lane).
  async_load_b128(xg + base + (size_t)tid * VEC, &xbuf[0][tid * VEC]);
  async_load_b128(tg + base + (size_t)tid * VEC, &tbuf[0][tid * VEC]);

  float sx = 0.0f, st = 0.0f, sxt = 0.0f, sbce = 0.0f;
  unsigned ncomp = 0;

  for (int c = 0; c < NCHUNKS; ++c) {
    const int buf = c & 1;
    if (c + 1 < NCHUNKS) {
      const size_t nb = base + (size_t)(c + 1) * CHUNK + (size_t)tid * VEC;
      async_load_b128(xg + nb, &xbuf[buf ^ 1][tid * VEC]);
      async_load_b128(tg + nb, &tbuf[buf ^ 1][tid * VEC]);
      WAIT_ASYNC(2);   // chunk c's 2 loads done; next stage stays in flight
    } else {
      WAIT_ASYNC(0);
    }
    __syncthreads();

    const float4 xv = *(const float4*)&xbuf[buf][tid * VEC];
    const float4 tv = *(const float4*)&tbuf[buf][tid * VEC];
    const float prev = (tid == 0) ? carry[c & 1] : xbuf[buf][tid * VEC - 1];

    auto acc1 = [&](float xx, float tt, float pv) {
      sx += xx;
      st += tt;
      sxt = __builtin_fmaf(xx, tt, sxt);
      float lx  = fmaxf(__logf(xx), LOG_CLAMP);          // max(log x, -100)
      float l1x = fmaxf(__logf(1.0f - xx), LOG_CLAMP);   // max(log1p(-x), -100)
      sbce -= tt * lx + (1.0f - tt) * l1x;
      if ((xx != 0.0f) & (pv == 0.0f)) ncomp++;          // run start
    };
    acc1(xv.x, tv.x, prev);
    acc1(xv.y, tv.y, xv.x);
    acc1(xv.z, tv.z, xv.y);
    acc1(xv.w, tv.w, xv.z);

    // Publish last element for the next chunk's tid==0 (double-buffered slot;
    // the next iteration's mid-barrier orders the write before the read).
    if (tid == TPB - 1) carry[(c + 1) & 1] = xv.w;
    __syncthreads();   // release buf before it is overwritten next iteration
  }

  // Block reduction in LDS.
  red0[tid] = sx; red1[tid] = st; red2[tid] = sxt; red3[tid] = sbce;
  redc[tid] = ncomp;
  __syncthreads();
  for (int s = TPB / 2; s > 0; s >>= 1) {
    if (tid < s) {
      red0[tid] += red0[tid + s];
      red1[tid] += red1[tid + s];
      red2[tid] += red2[tid + s];
      red3[tid] += red3[tid + s];
      redc[tid] += redc[tid + s];
    }
    __syncthreads();
  }
  if (tid == 0) {
    atomicAdd(&acc[0], red0[0]);
    atomicAdd(&acc[1], red1[0]);
    atomicAdd(&acc[2], red2[0]);
    atomicAdd(&acc[3], red3[0]);
    atomicAdd(reinterpret_cast<unsigned*>(acc) + 4, redc[0]);
  }
}

__global__ void dicebce_finalize(const float* __restrict__ acc,
                                 float* __restrict__ out) {
  if (threadIdx.x == 0 && blockIdx.x == 0) {
    const float sx   = acc[0];
    const float st   = acc[1];
    const float sxt  = acc[2];
    const float sbce = acc[3];
    const float nc   = (float)(reinterpret_cast<const unsigned*>(acc)[4]);
    const float bce  = sbce / (float)N_ELEMS;
    const float dice = 1.0f - (2.0f * sxt + 1.0f) / (sx + st + 1.0f);
    const float topo = fabsf(nc - 1.0f) / IMG_PIX;
    out[0] = 0.5f * bce + dice + topo;
  }
}

extern "C" void kernel_launch(void* const* d_in, const int* in_sizes, int n_in,
                              void* d_out, int out_size, void* d_ws, size_t ws_size,
                              hipStream_t stream) {
  (void)in_sizes; (void)n_in; (void)out_size; (void)ws_size;
  const float* x = (const float*)d_in[0];
  const float* t = (const float*)d_in[1];
  float* acc = (float*)d_ws;
  float* out = (float*)d_out;

  hipLaunchKernelGGL(dicebce_init_ws, dim3(1), dim3(32), 0, stream, acc);
  hipLaunchKernelGGL(dicebce_main, dim3(NBLOCKS), dim3(TPB), 0, stream, x, t, acc);
  hipLaunchKernelGGL(dicebce_finalize, dim3(1), dim3(32), 0, stream, acc, out);
}